// CDE_BCR_12850542150264
// MI455X (gfx1250) — compile-verified
//
#include <hip/hip_runtime.h>

// MI455X / gfx1250, wave32. All matmuls use V_WMMA_F32_16X16X4_F32:
// every GEMM in this pipeline has M=16 (batch or l-tile rows), and the
// dominant cost is streaming dense_W (100 MB, read once) -> bandwidth bound,
// so fp32 WMMA keeps full precision at no roofline cost. dense_W is staged
// into LDS by the Tensor Data Mover (double-buffered 16x128 chunks, LDS row
// padding by the TDM pad feature), overlapping DMA with WMMA.

typedef float v2f __attribute__((ext_vector_type(2)));
typedef float v8f __attribute__((ext_vector_type(8)));
typedef unsigned int v4u __attribute__((ext_vector_type(4)));
typedef int v4i __attribute__((ext_vector_type(4)));
typedef int v8i __attribute__((ext_vector_type(8)));

#define WMMA_F32(a, b, c) \
  __builtin_amdgcn_wmma_f32_16x16x4_f32(false, (a), false, (b), (short)0, (c), false, false)

#define SQH 0.70710678118654752440f

// ---------------------------------------------------------------------------
// TDM: load an nrows x 128 f32 tile (row stride 128 in memory) into LDS,
// padding +1 DWORD after every 128 DWORDs (LDS rows become 129 floats).
// D# layout per CDNA5 ISA 8.3/8.4 (group0 128b, group1 256b; groups 2/3 zero).
// Toolchain uses the 6-arg builtin: (v4u, v8i, v4i, v4i, v8i, cpol).
// ---------------------------------------------------------------------------
__device__ __forceinline__ void tdm_load_chunk(const float* gsrc,
                                               unsigned lds_byte, int nrows)
{
  unsigned long long ga = (unsigned long long)(uintptr_t)gsrc;
  v4u g0;
  g0.x = 1u;                                             // count=1 (valid user D#)
  g0.y = lds_byte;                                       // lds_addr
  g0.z = (unsigned)ga;                                   // global_addr[31:0]
  g0.w = ((unsigned)(ga >> 32) & 0x01FFFFFFu)            // global_addr[56:32]
         | 0x80000000u;                                  // type=2 ("image")
  v8i g1;
  g1[0] = (int)((2u << 16)        // data_size = 4 bytes
              | (1u << 20)        // pad_enable
              | (6u << 22));      // pad_interval: 2^(6+1)=128 DWORDs; amount=1 DWORD
  g1[1] = (int)(128u << 16);                 // tensor_dim0 = 128 (low16 in [63:48])
  g1[2] = (int)((unsigned)nrows << 16);      // tensor_dim1 low16 in [95:80]
  g1[3] = (int)(128u << 16);                 // tile_dim0 = 128 in [127:112]
  g1[4] = nrows;                             // tile_dim1 in [143:128]
  g1[5] = 128;                               // tensor_dim0_stride[31:0]
  g1[6] = 0;
  g1[7] = 0;
  v4i z4 = {0, 0, 0, 0};
  v8i z8 = {0, 0, 0, 0, 0, 0, 0, 0};
  __builtin_amdgcn_tensor_load_to_lds(g0, g1, z4, z4, z8, 0);
}

// ---------------------------------------------------------------------------
// Kernel 1: der, z=relu(seq@Wg), h=relu(z@Wh), v[b,k,l]=sum_D h[l,D,k]*der[l,D]
// ---------------------------------------------------------------------------
__global__ __launch_bounds__(256) void frontend_kernel(
    const float* __restrict__ seq, const float* __restrict__ coeffs,
    const float* __restrict__ tarr, const float* __restrict__ tstep,
    const float* __restrict__ Wg, const float* __restrict__ Wh,
    float* __restrict__ v_out)
{
  __shared__ float s_seq[16][65];
  __shared__ float s_der[16][65];
  __shared__ float s_z[16][33];
  __shared__ float s_vred[8][256];

  const int tid  = threadIdx.x;
  const int lane = tid & 31;
  const int wave = tid >> 5;
  const int b    = blockIdx.x >> 7;     // 128 l-tiles
  const int lt   = blockIdx.x & 127;
  const int l0   = lt * 16;

  for (int e = tid; e < 16 * 64; e += 256) {
    int r = e >> 6, c = e & 63;
    int l = l0 + r;
    s_seq[r][c] = seq[(b * 2048 + l) * 64 + c];
    float t = tarr[l];
    int i = (int)t; if (i < 0) i = 0; if (i > 2046) i = 2046;
    float dt = tstep[i + 1] - tstep[i];
    s_der[r][c] = (coeffs[(b * 2048 + i + 1) * 64 + c] -
                   coeffs[(b * 2048 + i) * 64 + c]) / dt;
  }
  __syncthreads();

  const int row  = lane & 15;
  const int hi   = lane >> 4;   // 0/1
  const int koff = hi * 2;
  const int m8   = hi * 8;

  if (wave == 0) {              // z = relu(seq @ Wg): M=16 K=64 N=32
    for (int nt = 0; nt < 2; ++nt) {
      v8f c = {};
      for (int ks = 0; ks < 16; ++ks) {
        int kb = ks * 4;
        v2f a, bm;
        a.x  = s_seq[row][kb + koff];
        a.y  = s_seq[row][kb + koff + 1];
        bm.x = Wg[(kb + koff) * 32 + nt * 16 + row];
        bm.y = Wg[(kb + koff + 1) * 32 + nt * 16 + row];
        c = WMMA_F32(a, bm, c);
      }
      for (int i = 0; i < 8; ++i) {
        float zv = c[i]; if (zv < 0.f) zv = 0.f;
        s_z[i + m8][nt * 16 + row] = zv;
      }
    }
  }
  __syncthreads();

  v2f az[8];                    // A-frags of z (16 x 32)
  for (int ks = 0; ks < 8; ++ks) {
    az[ks].x = s_z[row][ks * 4 + koff];
    az[ks].y = s_z[row][ks * 4 + koff + 1];
  }

  v8f vacc = {};
  for (int tt = 0; tt < 8; ++tt) {
    int T = wave * 8 + tt;      // D index
    v8f hc = {};
    for (int ks = 0; ks < 8; ++ks) {
      v2f bm;
      bm.x = Wh[(ks * 4 + koff) * 1024 + T * 16 + row];
      bm.y = Wh[(ks * 4 + koff + 1) * 1024 + T * 16 + row];
      hc = WMMA_F32(az[ks], bm, hc);
    }
    for (int i = 0; i < 8; ++i) {
      float hv = hc[i]; if (hv < 0.f) hv = 0.f;
      vacc[i] += hv * s_der[i + m8][T];
    }
  }
  for (int i = 0; i < 8; ++i) s_vred[wave][lane * 8 + i] = vacc[i];
  __syncthreads();
  if (wave == 0) {
    for (int i = 0; i < 8; ++i) {
      float s = 0.f;
      for (int w = 0; w < 8; ++w) s += s_vred[w][lane * 8 + i];
      v_out[(b * 16 + row) * 2048 + l0 + i + m8] = s;  // v[b, k=row, l]
    }
  }
}

// ---------------------------------------------------------------------------
// Kernel 2: 4-level Haar analysis of v rows (one block per (b,k) row)
// ---------------------------------------------------------------------------
__global__ __launch_bounds__(256) void haar_kernel(
    const float* __restrict__ v, float* __restrict__ det, float* __restrict__ app)
{
  __shared__ float bufA[2048];
  __shared__ float bufB[1024];
  const int bk = blockIdx.x, tid = threadIdx.x;
  for (int t = tid; t < 2048; t += 256) bufA[t] = v[bk * 2048 + t];
  __syncthreads();
  const int lvloff[4] = {0, 262144, 393216, 458752};
  float* src = bufA; float* dst = bufB;
  for (int lev = 0; lev < 4; ++lev) {
    int half = 1024 >> lev;
    for (int t = tid; t < half; t += 256) {
      float x0 = src[2 * t], x1 = src[2 * t + 1];
      float cA = (x0 + x1) * SQH, cD = (x0 - x1) * SQH;
      det[lvloff[lev] + bk * half + t] = cD;
      app[lvloff[lev] + bk * half + t] = cA;
      dst[t] = cA;
    }
    __syncthreads();
    float* tp = src; src = dst; dst = tp;
  }
}

// ---------------------------------------------------------------------------
// Kernel 3: dense chain, per (d,k): cur <- cur @ W[j]^T  (M=16,K=128,N=128) x3
// One wave per (d,k). dense_W (100 MB read once) is streamed by the TDM into
// a double-buffered LDS chunk (16 rows x 129 padded floats); WMMA on chunk nt
// overlaps the DMA of chunk nt+1. TENSORcnt gates consumption.
// ---------------------------------------------------------------------------
__global__ __launch_bounds__(32) void dense_kernel(
    const float* __restrict__ app, const float* __restrict__ dense_W,
    float* __restrict__ cur)
{
  __shared__ float s_cur[16][129];
  __shared__ float s_w[2][16 * 129];     // TDM-padded rows: 129 floats
  const int dk   = blockIdx.x;           // d_i*16 + k_i
  const int lane = threadIdx.x & 31;
  const int row = lane & 15, hi = lane >> 4, koff = hi * 2, m8 = hi * 8;

  const unsigned ldsw0 = (unsigned)(uintptr_t)&s_w[0][0];
  const unsigned ldsw1 = (unsigned)(uintptr_t)&s_w[1][0];

  for (int e = threadIdx.x; e < 16 * 128; e += 32) {
    int bb = e >> 7, q = e & 127;
    s_cur[bb][q] = app[458752 + (bb * 16 + (dk & 15)) * 128 + q];
  }
  __syncthreads();

  for (int j = 0; j < 3; ++j) {
    v2f a[32];
    for (int ks = 0; ks < 32; ++ks) {
      a[ks].x = s_cur[row][ks * 4 + koff];
      a[ks].y = s_cur[row][ks * 4 + koff + 1];
    }
    __syncthreads();
    const float* Wj = dense_W + (size_t)(j * 512 + dk) * 128 * 128;

    // prime chunk 0 (rows 0..15 of W[j])
    asm volatile("s_wait_dscnt 0" ::: "memory");   // LDS WAR guard vs prev j
    tdm_load_chunk(Wj, ldsw0, 16);

    for (int nt = 0; nt < 8; ++nt) {
      if (nt < 7) {
        asm volatile("s_wait_dscnt 0" ::: "memory"); // WAR on buf[(nt+1)&1]
        tdm_load_chunk(Wj + (size_t)(nt + 1) * 16 * 128,
                       ((nt + 1) & 1) ? ldsw1 : ldsw0, 16);
        __builtin_amdgcn_s_wait_tensorcnt(1);        // chunk nt landed
      } else {
        __builtin_amdgcn_s_wait_tensorcnt(0);
      }
      const float* wbuf = &s_w[nt & 1][0];
      v8f c = {};
      for (int ks = 0; ks < 32; ++ks) {
        v2f bm;                           // B[q][t] = W[t][q]; row = t local
        bm.x = wbuf[row * 129 + ks * 4 + koff];
        bm.y = wbuf[row * 129 + ks * 4 + koff + 1];
        c = WMMA_F32(a[ks], bm, c);
      }
      for (int i = 0; i < 8; ++i) s_cur[i + m8][nt * 16 + row] = c[i];
    }
    __syncthreads();
  }
  for (int e = threadIdx.x; e < 16 * 128; e += 32) {
    int bb = e >> 7, t = e & 127;
    cur[(size_t)(dk * 16 + bb) * 2048 + t] = s_cur[bb][t];
  }
}

// ---------------------------------------------------------------------------
// Kernel 4: locally-connected conv (kernel 5, 2ch->2ch, 8 segments) + relu
// ---------------------------------------------------------------------------
__global__ __launch_bounds__(256) void lc_kernel(
    const float* __restrict__ src,
    const float* __restrict__ det, const float* __restrict__ app,
    const float* __restrict__ lc_w, const float* __restrict__ lc_b,
    float* __restrict__ dst, int lev, int j, int Ll, int lvloff)
{
  const int dk = blockIdx.y, d_i = dk >> 4, k_i = dk & 15;
  int gid = blockIdx.x * 256 + threadIdx.x;
  if (gid >= 16 * 2 * Ll) return;
  int t = gid % Ll;
  int o = (gid / Ll) & 1;
  int b = gid / (2 * Ll);
  int seg = t / (Ll >> 3); if (seg > 7) seg = 7;
  size_t pbase = (((size_t)(lev * 3 + j) * 32 + d_i) * 16 + k_i) * 2 + o;
  const float* wbase = lc_w + pbase * 2 * 8 * 5;
  float acc = lc_b[pbase * 8 + seg];
  for (int ii = 0; ii < 2; ++ii) {
    const float* wv = wbase + ((size_t)ii * 8 + seg) * 5;
    for (int f = 0; f < 5; ++f) {
      int xl = t - 2 + f;
      float x = 0.f;
      if (xl >= 0 && xl < Ll) {
        if (j == 0) {
          const float* s0 = (ii == 0) ? det : app;   // stack: 0=detail, 1=approx
          x = s0[lvloff + (b * 16 + k_i) * Ll + xl];
        } else {
          x = src[(((size_t)dk * 16 + b) * 2 + ii) * Ll + xl];
        }
      }
      acc += wv[f] * x;
    }
  }
  if (acc < 0.f) acc = 0.f;
  dst[(((size_t)dk * 16 + b) * 2 + o) * Ll + t] = acc;
}

// ---------------------------------------------------------------------------
// Kernel 5: X[...,1,:] += cur; cur = haar_rec(X1, X0) (length doubles)
// ---------------------------------------------------------------------------
__global__ __launch_bounds__(256) void recon_kernel(
    const float* __restrict__ chi, const float* __restrict__ curin,
    float* __restrict__ curout, int Ll)
{
  const int dk = blockIdx.y;
  int gid = blockIdx.x * 256 + threadIdx.x;
  if (gid >= 16 * Ll) return;
  int t = gid % Ll, b = gid / Ll;
  size_t rb = (size_t)dk * 16 + b;
  float cD = chi[(rb * 2 + 0) * Ll + t];
  float cA = chi[(rb * 2 + 1) * Ll + t] + curin[rb * 2048 + t];
  curout[rb * 2048 + 2 * t]     = (cA + cD) * SQH;
  curout[rb * 2048 + 2 * t + 1] = (cA - cD) * SQH;
}

// ---------------------------------------------------------------------------
// Kernel 6: out = sum_k cur; U = out^T @ Wrev  (M=16 l-rows, K=32, N=64)
// ---------------------------------------------------------------------------
__global__ __launch_bounds__(256) void output_kernel(
    const float* __restrict__ cur, const float* __restrict__ Wrev,
    float* __restrict__ U)
{
  __shared__ float s_s[16][33];
  const int tid = threadIdx.x;
  const int b = blockIdx.x >> 7, lt = blockIdx.x & 127;
  const int l0 = lt * 16;
  for (int e = tid; e < 16 * 32; e += 256) {
    int r = e & 15, d_i = e >> 4;
    float s = 0.f;
    for (int kk = 0; kk < 16; ++kk)
      s += cur[((size_t)(d_i * 16 + kk) * 16 + b) * 2048 + l0 + r];
    s_s[r][d_i] = s;
  }
  __syncthreads();
  const int lane = tid & 31, wave = tid >> 5;
  const int row = lane & 15, hi = lane >> 4, koff = hi * 2, m8 = hi * 8;
  if (wave < 4) {               // wave-uniform: EXEC all-1s inside
    int nt = wave;
    v8f c = {};
    for (int ks = 0; ks < 8; ++ks) {
      v2f a, bm;
      a.x  = s_s[row][ks * 4 + koff];
      a.y  = s_s[row][ks * 4 + koff + 1];
      bm.x = Wrev[(ks * 4 + koff) * 64 + nt * 16 + row];
      bm.y = Wrev[(ks * 4 + koff + 1) * 64 + nt * 16 + row];
      c = WMMA_F32(a, bm, c);
    }
    for (int i = 0; i < 8; ++i)
      U[((size_t)b * 2048 + l0 + i + m8) * 64 + nt * 16 + row] = c[i];
  }
}

// ---------------------------------------------------------------------------
extern "C" void kernel_launch(void* const* d_in, const int* in_sizes, int n_in,
                              void* d_out, int out_size, void* d_ws, size_t ws_size,
                              hipStream_t stream)
{
  const float* seq     = (const float*)d_in[0];
  const float* coeffs  = (const float*)d_in[1];
  const float* tarr    = (const float*)d_in[2];
  const float* tstep   = (const float*)d_in[3];
  const float* Wg      = (const float*)d_in[4];
  const float* Wh      = (const float*)d_in[5];
  const float* dense_W = (const float*)d_in[6];
  const float* lc_w    = (const float*)d_in[7];
  const float* lc_b    = (const float*)d_in[8];
  const float* Wrev    = (const float*)d_in[9];
  float* U  = (float*)d_out;
  float* ws = (float*)d_ws;

  float* v    = ws;                      // B*k*L            = 524288
  float* det  = v + 524288;              // pyramid          = 491520
  float* app  = det + 491520;            // pyramid          = 491520
  float* curA = app + 491520;            // d*k*B*2048       = 16777216
  float* curB = curA + 16777216;
  float* chiA = curB + 16777216;         // d*k*B*2*1024     = 16777216
  float* chiB = chiA + 16777216;

  frontend_kernel<<<dim3(16 * 128), 256, 0, stream>>>(seq, coeffs, tarr, tstep, Wg, Wh, v);
  haar_kernel<<<dim3(256), 256, 0, stream>>>(v, det, app);
  dense_kernel<<<dim3(512), 32, 0, stream>>>(app, dense_W, curA);

  const int lvloff[4] = {0, 262144, 393216, 458752};
  float* curin = curA; float* curout = curB;
  for (int lev = 3; lev >= 0; --lev) {
    int Ll = 2048 >> (lev + 1);
    float* chin = chiB; float* chout = chiA;   // j==0 ignores chin
    for (int j = 0; j < 3; ++j) {
      dim3 grid((16 * 2 * Ll + 255) / 256, 512);
      lc_kernel<<<grid, 256, 0, stream>>>(chin, det, app, lc_w, lc_b, chout,
                                          lev, j, Ll, lvloff[lev]);
      float* tp = chin; chin = chout; chout = tp;
    }
    dim3 rgrid((16 * Ll + 255) / 256, 512);
    recon_kernel<<<rgrid, 256, 0, stream>>>(chin, curin, curout, Ll);
    float* tp = curin; curin = curout; curout = tp;
  }
  output_kernel<<<dim3(16 * 128), 256, 0, stream>>>(curin, Wrev, U);
}